// GAT_18047452578509
// MI455X (gfx1250) — compile-verified
//
#include <hip/hip_runtime.h>
#include <math.h>

// ---------------------------------------------------------------------------
// GAT (2-layer, PyG-style) for MI455X / gfx1250, wave32.
// - GEMM1: async-load-to-LDS staged panels + V_WMMA_F32_16X16X4_F32
// - GEMM2: direct-global fp32 WMMA (N=32 panel too narrow for LDS staging)
// - Edge phase: native f32 global atomics (L2-resident; 192MB L2 >> 56MB set)
// ---------------------------------------------------------------------------

typedef float v2f __attribute__((ext_vector_type(2)));
typedef float v8f __attribute__((ext_vector_type(8)));

#define NN      10000
#define EE      320000
#define E2      (EE + NN)      // edges + self loops
#define INDIM   128
#define H1DIM   512            // 8 heads * 64
#define HEADS1  8
#define D1      64
#define OUTD    32
#define NEG_SLOPE 0.2f

#define CEILDIV(a, b) (((a) + (b) - 1) / (b))

// Monotone order-preserving float<->uint encoding for atomic max on u32.
__device__ __forceinline__ unsigned fenc(float f) {
  unsigned u = __float_as_uint(f);
  return (u & 0x80000000u) ? ~u : (u | 0x80000000u);
}
__device__ __forceinline__ float fdec(unsigned u) {
  unsigned b = (u & 0x80000000u) ? (u ^ 0x80000000u) : ~u;
  return __uint_as_float(b);
}

// Generic pointer to __shared__ -> 32-bit LDS byte offset (AS3 ptrtoint).
__device__ __forceinline__ unsigned lds_off(const void* p) {
  return (unsigned)(__UINTPTR_TYPE__)(__attribute__((address_space(3))) const void*)p;
}

// GLOBAL_LOAD_ASYNC_TO_LDS_B128, GV mode: per-lane LDS offset + 64-bit vaddr.
// Tracked by ASYNCcnt (CDNA5 ISA).
__device__ __forceinline__ void async_copy_b128(unsigned lds, unsigned long long gaddr) {
  asm volatile("global_load_async_to_lds_b128 %0, %1, off"
               :: "v"(lds), "v"(gaddr) : "memory");
}
__device__ __forceinline__ void wait_asynccnt0() {
  asm volatile("s_wait_asynccnt 0" ::: "memory");
}

__global__ void zero_kernel(float* __restrict__ p, int n) {
  int i = blockIdx.x * blockDim.x + threadIdx.x;
  if (i < n) p[i] = 0.0f;
}

// ---------------------------------------------------------------------------
// LDS-staged fp32 WMMA GEMM: C[M,N] = A[M,K] @ B[K,N], row-major.
// One 256-thread block owns a 16(M) x 128(N) C panel (8 waves x one 16x16
// tile each). K consumed in 32-wide chunks: A(16x32)+B(32x128) staged into
// LDS via async b128 copies, then WMMA fed from LDS.
// Requires M%16==0, N%128==0, K%32==0 (GEMM1: 10000/512/128 — exact fit).
// ---------------------------------------------------------------------------
#define KC 32
__global__ void __launch_bounds__(256)
wmma_gemm_f32_lds(const float* __restrict__ A, const float* __restrict__ B,
                  float* __restrict__ C, int M, int N, int K) {
  __shared__ float Asm[16 * KC];    // 2 KB
  __shared__ float Bsm[KC * 128];   // 16 KB
  int npanels = N >> 7;
  int m0 = (blockIdx.x / npanels) << 4;
  int n0 = (blockIdx.x % npanels) << 7;
  int tid  = threadIdx.x;
  int wave = tid >> 5;
  int lane = tid & 31;
  int l15 = lane & 15, hi = lane >> 4;
  int bn = (wave << 4) + l15;       // this lane's B column within the panel

  // async-copy coordinates: A moved by 128 threads (1x b128), B by 256 (4x).
  int arow = tid >> 3, acol = (tid & 7) << 2;        // A: 16 rows x 8 xb128
  unsigned a_lds = lds_off(&Asm[arow * KC + acol]);
  unsigned b_lds = lds_off(&Bsm[0]);

  v8f acc = {};
  for (int kc = 0; kc < K; kc += KC) {
    __syncthreads();                 // previous chunk fully consumed
    if (tid < 128)
      async_copy_b128(a_lds,
          (unsigned long long)(A + (size_t)(m0 + arow) * K + kc + acol));
#pragma unroll
    for (int it = 0; it < 4; ++it) { // B: 32 rows x 32 xb128
      int idx = tid + (it << 8);
      int brow = idx >> 5, bcol = (idx & 31) << 2;
      async_copy_b128(b_lds + ((unsigned)(brow * 128 + bcol) << 2),
          (unsigned long long)(B + (size_t)(kc + brow) * N + n0 + bcol));
    }
    wait_asynccnt0();                // this wave's async copies landed in LDS
    __syncthreads();                 // ... and everyone else's too
#pragma unroll
    for (int k0 = 0; k0 < KC; k0 += 4) {
      int ka = k0 + (hi << 1);       // lanes 16-31 hold K=+2..+3 (ISA §7.12.2)
      v2f a, b;
      a.x = Asm[l15 * KC + ka];
      a.y = Asm[l15 * KC + ka + 1];
      b.x = Bsm[ka * 128 + bn];
      b.y = Bsm[(ka + 1) * 128 + bn];
      acc = __builtin_amdgcn_wmma_f32_16x16x4_f32(
          false, a, false, b, (short)0, acc, false, false);
    }
  }
  // C/D layout: VGPR v -> M = m0 + v + 8*hi, N = n0 + bn
  float* Cp = C + (size_t)(m0 + (hi << 3)) * N + n0 + bn;
#pragma unroll
  for (int v = 0; v < 8; ++v) Cp[(size_t)v * N] = acc[v];
}

// ---------------------------------------------------------------------------
// Direct-global fp32 WMMA GEMM (used for the narrow GEMM2, N=32).
// One wave32 computes one 16x16 C tile via K/4 v_wmma_f32_16x16x4_f32 ops.
// ---------------------------------------------------------------------------
__global__ void wmma_gemm_f32(const float* __restrict__ A,
                              const float* __restrict__ B,
                              float* __restrict__ C,
                              int M, int N, int K) {
  int wave = (blockIdx.x * blockDim.x + threadIdx.x) >> 5;
  int lane = threadIdx.x & 31;
  int ntiles = N >> 4;
  int tiles  = (M >> 4) * ntiles;
  if (wave >= tiles) return;
  int m0 = (wave / ntiles) << 4;
  int n0 = (wave % ntiles) << 4;
  int l15 = lane & 15;
  int hi  = lane >> 4;

  const float* Arow = A + (size_t)(m0 + l15) * K;
  int bcol = n0 + l15;
  v8f acc = {};
  for (int k0 = 0; k0 < K; k0 += 4) {
    int ka = k0 + (hi << 1);
    v2f a, b;
    a.x = Arow[ka];
    a.y = Arow[ka + 1];
    b.x = B[(size_t)ka * N + bcol];
    b.y = B[(size_t)(ka + 1) * N + bcol];
    acc = __builtin_amdgcn_wmma_f32_16x16x4_f32(
        false, a, false, b, (short)0, acc, false, false);
  }
  float* Cp = C + (size_t)(m0 + (hi << 3)) * N + n0 + l15;
#pragma unroll
  for (int v = 0; v < 8; ++v) Cp[(size_t)v * N] = acc[v];
}

// Per-(node,head) attention dots: as[i] = h[i,:]·a_src[head], ad likewise.
__global__ void alpha_kernel(const float* __restrict__ h,
                             const float* __restrict__ a_src,
                             const float* __restrict__ a_dst,
                             float* __restrict__ as_out,
                             float* __restrict__ ad_out,
                             int nNodes, int heads, int d) {
  int i = blockIdx.x * blockDim.x + threadIdx.x;
  if (i >= nNodes * heads) return;
  int head = i % heads;
  const float* hp  = h + (size_t)i * d;
  const float* asp = a_src + head * d;
  const float* adp = a_dst + head * d;
  float s = 0.f, t = 0.f;
  for (int j = 0; j < d; ++j) { float v = hp[j]; s += v * asp[j]; t += v * adp[j]; }
  as_out[i] = s;
  ad_out[i] = t;
}

// Edge pass 1: leaky-relu logits + segment max (encoded u32 atomicMax).
__global__ void logits_kernel(const int* __restrict__ src,
                              const int* __restrict__ dst,
                              const float* __restrict__ as_,
                              const float* __restrict__ ad_,
                              float* __restrict__ logits,
                              unsigned* __restrict__ mEnc,
                              int nEdges, int nBase, int heads) {
  int e = blockIdx.x * blockDim.x + threadIdx.x;
  if (e >= nEdges) return;
  int s  = (e < nBase) ? src[e] : (e - nBase);
  int dn = (e < nBase) ? dst[e] : (e - nBase);
  for (int h = 0; h < heads; ++h) {
    float l = as_[s * heads + h] + ad_[dn * heads + h];
    l = (l > 0.f) ? l : NEG_SLOPE * l;
    logits[(size_t)e * heads + h] = l;
    atomicMax(&mEnc[dn * heads + h], fenc(l));
  }
}

// Edge pass 2: e = exp(logit - m[dst]); segment sum into denom.
__global__ void exp_kernel(const int* __restrict__ dst,
                           float* __restrict__ logits,
                           const unsigned* __restrict__ mEnc,
                           float* __restrict__ denom,
                           int nEdges, int nBase, int heads) {
  int e = blockIdx.x * blockDim.x + threadIdx.x;
  if (e >= nEdges) return;
  int dn = (e < nBase) ? dst[e] : (e - nBase);
  for (int h = 0; h < heads; ++h) {
    float m = fdec(mEnc[dn * heads + h]);
    float v = expf(logits[(size_t)e * heads + h] - m);
    logits[(size_t)e * heads + h] = v;
    atomicAdd(&denom[dn * heads + h], v);
  }
}

// Edge pass 3: one wave32 per (edge, head); lanes stripe the d feature dims.
__global__ void agg_kernel(const int* __restrict__ src,
                           const int* __restrict__ dst,
                           const float* __restrict__ evals,
                           const float* __restrict__ denom,
                           const float* __restrict__ h,
                           float* __restrict__ out,
                           int nEdges, int nBase, int heads, int d) {
  int wave = (blockIdx.x * blockDim.x + threadIdx.x) >> 5;
  int lane = threadIdx.x & 31;
  if (wave >= nEdges * heads) return;
  int e    = wave / heads;
  int head = wave % heads;
  int s  = (e < nBase) ? src[e] : (e - nBase);
  int dn = (e < nBase) ? dst[e] : (e - nBase);
  const float* hp = h + ((size_t)s * heads + head) * d;
  __builtin_prefetch(hp + lane, 0, 0);    // global_prefetch_b8 on gfx1250
  float alpha = evals[(size_t)e * heads + head] / denom[dn * heads + head];
  float* op = out + ((size_t)dn * heads + head) * d;
  for (int j = lane; j < d; j += 32) atomicAdd(&op[j], alpha * hp[j]);
}

// In-place x = elu(x + b[i % dim])
__global__ void elu_bias_kernel(float* __restrict__ x, const float* __restrict__ b,
                                int n, int dim) {
  int i = blockIdx.x * blockDim.x + threadIdx.x;
  if (i >= n) return;
  float v = x[i] + b[i % dim];
  x[i] = (v > 0.f) ? v : (expf(v) - 1.0f);
}

__global__ void add_bias_kernel(float* __restrict__ x, const float* __restrict__ b,
                                int n, int dim) {
  int i = blockIdx.x * blockDim.x + threadIdx.x;
  if (i >= n) return;
  x[i] += b[i % dim];
}

// ---------------------------------------------------------------------------
extern "C" void kernel_launch(void* const* d_in, const int* in_sizes, int n_in,
                              void* d_out, int out_size, void* d_ws, size_t ws_size,
                              hipStream_t stream) {
  const float* x      = (const float*)d_in[0];
  const int*   ei     = (const int*)  d_in[1];   // [2, E] int32
  const float* W1     = (const float*)d_in[2];
  const float* a1_src = (const float*)d_in[3];
  const float* a1_dst = (const float*)d_in[4];
  const float* b1     = (const float*)d_in[5];
  const float* W2     = (const float*)d_in[6];
  const float* a2_src = (const float*)d_in[7];
  const float* a2_dst = (const float*)d_in[8];
  const float* b2     = (const float*)d_in[9];
  const int* srcI = ei;
  const int* dstI = ei + EE;
  float* out = (float*)d_out;

  // --- workspace layout (floats); zeroed buffers kept contiguous ---
  float* ws = (float*)d_ws;
  float*    h1      = ws;                 ws += (size_t)NN * H1DIM;
  float*    logits1 = ws;                 ws += (size_t)E2 * HEADS1;
  float*    as1     = ws;                 ws += (size_t)NN * HEADS1;
  float*    ad1     = ws;                 ws += (size_t)NN * HEADS1;
  unsigned* m1      = (unsigned*)ws;      ws += (size_t)NN * HEADS1;  // zero A
  float*    den1    = ws;                 ws += (size_t)NN * HEADS1;
  float*    agg1    = ws;                 ws += (size_t)NN * H1DIM;   // zero A end
  float*    z       = ws;                 ws += (size_t)NN * OUTD;
  float*    as2     = ws;                 ws += NN;
  float*    ad2     = ws;                 ws += NN;
  float*    logits2 = ws;                 ws += E2;
  unsigned* m2      = (unsigned*)ws;      ws += NN;                   // zero B
  float*    den2    = ws;                 ws += NN;

  const int B = 256;

  // 0) zero init: [m1|den1|agg1], [m2|den2], out   (u32 0 == f32 0 bits)
  {
    int nA = NN * HEADS1 * 2 + NN * H1DIM;
    zero_kernel<<<CEILDIV(nA, B), B, 0, stream>>>((float*)m1, nA);
    int nB2 = NN * 2;
    zero_kernel<<<CEILDIV(nB2, B), B, 0, stream>>>((float*)m2, nB2);
    zero_kernel<<<CEILDIV(NN * OUTD, B), B, 0, stream>>>(out, NN * OUTD);
  }

  // ---------------- Layer 1 ----------------
  // 1) h1 = x @ W1   (10000x128x512): async-LDS staged WMMA
  {
    int blocks = (NN / 16) * (H1DIM / 128);          // 625 * 4 = 2500
    wmma_gemm_f32_lds<<<blocks, B, 0, stream>>>(x, W1, h1, NN, H1DIM, INDIM);
  }
  // 2) attention dots
  alpha_kernel<<<CEILDIV(NN * HEADS1, B), B, 0, stream>>>(h1, a1_src, a1_dst, as1, ad1,
                                                          NN, HEADS1, D1);
  // 3-5) edge softmax + aggregation
  logits_kernel<<<CEILDIV(E2, B), B, 0, stream>>>(srcI, dstI, as1, ad1, logits1, m1,
                                                  E2, EE, HEADS1);
  exp_kernel<<<CEILDIV(E2, B), B, 0, stream>>>(dstI, logits1, m1, den1, E2, EE, HEADS1);
  {
    long long thr = (long long)E2 * HEADS1 * 32;
    agg_kernel<<<(unsigned)CEILDIV(thr, (long long)B), B, 0, stream>>>(
        srcI, dstI, logits1, den1, h1, agg1, E2, EE, HEADS1, D1);
  }
  // 6) h2in = elu(agg1 + b1), in place
  elu_bias_kernel<<<CEILDIV(NN * H1DIM, B), B, 0, stream>>>(agg1, b1, NN * H1DIM, H1DIM);

  // ---------------- Layer 2 ----------------
  // 7) z = h2in @ W2   (10000x512x32): direct-global WMMA
  {
    int waves = (NN / 16) * (OUTD / 16);             // 1250
    wmma_gemm_f32<<<CEILDIV(waves * 32, B), B, 0, stream>>>(agg1, W2, z, NN, OUTD, H1DIM);
  }
  alpha_kernel<<<CEILDIV(NN, B), B, 0, stream>>>(z, a2_src, a2_dst, as2, ad2, NN, 1, OUTD);
  logits_kernel<<<CEILDIV(E2, B), B, 0, stream>>>(srcI, dstI, as2, ad2, logits2, m2,
                                                  E2, EE, 1);
  exp_kernel<<<CEILDIV(E2, B), B, 0, stream>>>(dstI, logits2, m2, den2, E2, EE, 1);
  {
    long long thr = (long long)E2 * 32;
    agg_kernel<<<(unsigned)CEILDIV(thr, (long long)B), B, 0, stream>>>(
        srcI, dstI, logits2, den2, z, out, E2, EE, 1, OUTD);
  }
  add_bias_kernel<<<CEILDIV(NN * OUTD, B), B, 0, stream>>>(out, b2, NN * OUTD, OUTD);
}